// BrainGNN_46308337385708
// MI455X (gfx1250) — compile-verified
//
#include <hip/hip_runtime.h>
#include <hip/hip_bf16.h>

#define H_DIM 64
#define BN_EPS 1e-5f
#define ROWS_PER_BLOCK 256
#define RT_PER_BLOCK (ROWS_PER_BLOCK / 16)

typedef float v2f __attribute__((ext_vector_type(2)));
typedef float v8f __attribute__((ext_vector_type(8)));
typedef int   v4i __attribute__((ext_vector_type(4)));

#define AS_GLOBAL __attribute__((address_space(1)))
#define AS_LOCAL  __attribute__((address_space(3)))

static __host__ __device__ inline int cdiv_i(int a, int b) { return (a + b - 1) / b; }

// ---------------------------------------------------------------------------
// Degree / normalization
// ---------------------------------------------------------------------------
__global__ void gcn_deg_init(float* __restrict__ deg, int n) {
    int i = blockIdx.x * blockDim.x + threadIdx.x;
    if (i < n) deg[i] = 1.0f;  // self-loop contribution
}

__global__ void gcn_deg_count(const int* __restrict__ dstI, float* __restrict__ deg, int e) {
    int i = blockIdx.x * blockDim.x + threadIdx.x;
    if (i < e) atomicAdd(&deg[dstI[i]], 1.0f);
}

__global__ void gcn_deg_rsqrt(float* __restrict__ deg, int n) {
    int i = blockIdx.x * blockDim.x + threadIdx.x;
    if (i < n) deg[i] = rsqrtf(deg[i]);  // deg >= 1 always
}

__global__ void gcn_edge_norm(const int* __restrict__ srcI, const int* __restrict__ dstI,
                              const float* __restrict__ dinv, float* __restrict__ norm, int e) {
    int i = blockIdx.x * blockDim.x + threadIdx.x;
    if (i < e) norm[i] = dinv[srcI[i]] * dinv[dstI[i]];
}

// ---------------------------------------------------------------------------
// Dense GEMM: out[n x 64] = A[n x K] @ W[K x 64] via V_WMMA_F32_16X16X4_F32.
// Block = 256 rows x 64 cols, 4 waves (one 16-col tile each).
//  - A tile (256 x K) staged in LDS once per block (async-to-LDS),
//    padded row stride (K+4 dwords) to spread the 16 row-lanes over banks.
//  - B fragments for the whole K held in registers (K/4 x v2f), reused for
//    all 16 row-tiles.
// ---------------------------------------------------------------------------
template <int K>
__global__ __launch_bounds__(128) void gcn_gemm_wmma_k(const float* __restrict__ A,
                                                       const float* __restrict__ W,
                                                       float* __restrict__ out,
                                                       int nrows) {
    constexpr int KP   = K + 4;       // padded LDS row stride (dwords)
    constexpr int SEGS = K / 4;       // float4 segments per row
    extern __shared__ float sA[];     // ROWS_PER_BLOCK * KP floats

    const int tid   = threadIdx.x;
    const int lane  = tid & 31;
    const int wave  = tid >> 5;       // 0..3 -> 16-col tile
    const int half  = lane >> 4;      // 0: lanes 0-15, 1: lanes 16-31
    const int l15   = lane & 15;
    const int m_blk = blockIdx.x * ROWS_PER_BLOCK;
    const int n_base = wave << 4;
    const int kb     = half << 1;     // K-pair offset within a 4-step: 0 or 2

    // ---- Stage A tile (ROWS_PER_BLOCK x K) into LDS ----
    for (int idx = tid; idx < ROWS_PER_BLOCK * SEGS; idx += 128) {
        int row = idx / SEGS;
        int seg = idx - row * SEGS;
        int gr  = m_blk + row;
        if (gr >= nrows) gr = nrows - 1;            // clamp for tail block
        const float* gp = A + (size_t)gr * K + seg * 4;
        float*       lp = sA + row * KP + seg * 4;
#if defined(__gfx1250__) && __has_builtin(__builtin_amdgcn_global_load_async_to_lds_b128)
        __builtin_amdgcn_global_load_async_to_lds_b128(
            (AS_GLOBAL v4i*)gp, (AS_LOCAL v4i*)lp, 0, 0);
#else
        *(float4*)lp = *(const float4*)gp;
#endif
    }
#if defined(__gfx1250__) && __has_builtin(__builtin_amdgcn_global_load_async_to_lds_b128)
    asm volatile("s_wait_asynccnt 0" ::: "memory");
#endif
    __syncthreads();

    // ---- Preload all B fragments for this wave's col tile into registers ----
    v2f breg[SEGS];
    const float* bp = W + n_base + l15;
#pragma unroll
    for (int j = 0; j < SEGS; ++j) {
        int k0 = j * 4;
        v2f b;
        b.x = bp[(k0 + kb) * H_DIM];                // W[k0+kb][n]
        b.y = bp[(k0 + kb + 1) * H_DIM];            // W[k0+kb+1][n]
        breg[j] = b;
    }

    // ---- Row-tile loop: A fragments from LDS, B from registers ----
    for (int rt = 0; rt < RT_PER_BLOCK; ++rt) {
        const float* arow = sA + (rt * 16 + l15) * KP + kb;
        v8f c = {};
#pragma unroll
        for (int j = 0; j < SEGS; ++j) {
            v2f a = *(const v2f*)(arow + j * 4);    // A[m][k0+kb], A[m][k0+kb+1]
#if __has_builtin(__builtin_amdgcn_wmma_f32_16x16x4_f32)
            c = __builtin_amdgcn_wmma_f32_16x16x4_f32(false, a, false, breg[j],
                                                      (short)0, c, false, false);
#else
            c[0] += a.x * breg[j].x;  // host-pass placeholder (never executed)
#endif
        }
        // D layout: VGPR r -> row (rt*16 + half*8 + r), col n_base + l15
        const int mout = m_blk + rt * 16 + (half << 3);
        float* op = out + (size_t)mout * H_DIM + n_base + l15;
#pragma unroll
        for (int r = 0; r < 8; ++r)
            if (mout + r < nrows) op[(size_t)r * H_DIM] = c[r];
    }
}

// ---------------------------------------------------------------------------
// Aggregation: agg = selfloop-init, then atomic scatter over edges.
// 16 threads / row (or edge), float4 per thread.
// ---------------------------------------------------------------------------
__global__ void gcn_selfloop_init(const float* __restrict__ t, const float* __restrict__ dinv,
                                  float* __restrict__ agg, int n) {
    int tid = blockIdx.x * blockDim.x + threadIdx.x;
    if (tid >= n * 16) return;
    int i  = tid >> 4;
    int f4 = (tid & 15) << 2;
    float d  = dinv[i];
    float nm = d * d;                  // self-loop norm = dinv[i]^2
    float4 v = *(const float4*)(t + ((size_t)i << 6) + f4);
    float4* ap = (float4*)(agg + ((size_t)i << 6) + f4);
    float4 o; o.x = v.x * nm; o.y = v.y * nm; o.z = v.z * nm; o.w = v.w * nm;
    *ap = o;
}

__global__ void gcn_edge_scatter(const int* __restrict__ srcI, const int* __restrict__ dstI,
                                 const float* __restrict__ norm,
                                 const float* __restrict__ t, float* __restrict__ agg, int e) {
    int tid = blockIdx.x * blockDim.x + threadIdx.x;   // e*16 = 25.6M < 2^31
    if (tid >= e * 16) return;
    int ed = tid >> 4;
    int f4 = (tid & 15) << 2;
    int s = srcI[ed];
    int d = dstI[ed];
    float nm = norm[ed];
    float4 v = *(const float4*)(t + ((size_t)s << 6) + f4);
    float* ap = agg + ((size_t)d << 6) + f4;
    atomicAdd(ap + 0, v.x * nm);
    atomicAdd(ap + 1, v.y * nm);
    atomicAdd(ap + 2, v.z * nm);
    atomicAdd(ap + 3, v.w * nm);
}

// Fused bias + eval-mode BatchNorm + ReLU
__global__ void gcn_post_bn_relu(const float* __restrict__ agg,
                                 const float* __restrict__ b, const float* __restrict__ g,
                                 const float* __restrict__ beta, const float* __restrict__ m,
                                 const float* __restrict__ v, float* __restrict__ h, int n) {
    int tid = blockIdx.x * blockDim.x + threadIdx.x;
    if (tid >= n * H_DIM) return;
    int f = tid & (H_DIM - 1);
    float scale = g[f] * rsqrtf(v[f] + BN_EPS);
    float val = (agg[tid] + b[f] - m[f]) * scale + beta[f];
    h[tid] = fmaxf(val, 0.0f);
}

// ---------------------------------------------------------------------------
// Pooling + MLP head
// ---------------------------------------------------------------------------
__global__ void gcn_zero(float* __restrict__ p, int n) {
    int i = blockIdx.x * blockDim.x + threadIdx.x;
    if (i < n) p[i] = 0.0f;
}

__global__ void gcn_pool(const float* __restrict__ h, const int* __restrict__ batch,
                         float* __restrict__ pooled, int n) {
    int tid = blockIdx.x * blockDim.x + threadIdx.x;
    if (tid >= n * 16) return;
    int i  = tid >> 4;
    int f4 = (tid & 15) << 2;
    int g = batch[i];
    float4 v = *(const float4*)(h + ((size_t)i << 6) + f4);
    float* p = pooled + ((size_t)g << 6) + f4;
    atomicAdd(p + 0, v.x);
    atomicAdd(p + 1, v.y);
    atomicAdd(p + 2, v.z);
    atomicAdd(p + 3, v.w);
}

__global__ void gcn_mlp1(const float* __restrict__ pooled, const float* __restrict__ w,
                         const float* __restrict__ b, float* __restrict__ mid, int G) {
    int tid = blockIdx.x * blockDim.x + threadIdx.x;   // G*32
    if (tid >= G * 32) return;
    int g = tid >> 5;
    int j = tid & 31;
    float acc = b[j];
    const float* pr = pooled + ((size_t)g << 6);
#pragma unroll 8
    for (int k = 0; k < 64; ++k) acc += pr[k] * w[k * 32 + j];
    mid[tid] = fmaxf(acc, 0.0f);
}

__global__ void gcn_mlp2(const float* __restrict__ mid, const float* __restrict__ w,
                         const float* __restrict__ b, float* __restrict__ out, int G) {
    int tid = blockIdx.x * blockDim.x + threadIdx.x;   // G*2
    if (tid >= G * 2) return;
    int g = tid >> 1;
    int c = tid & 1;
    float acc = b[c];
    const float* mr = mid + (size_t)g * 32;
#pragma unroll
    for (int j = 0; j < 32; ++j) acc += mr[j] * w[j * 2 + c];
    out[tid] = acc;
}

// ---------------------------------------------------------------------------
// Host launcher
// ---------------------------------------------------------------------------
extern "C" void kernel_launch(void* const* d_in, const int* in_sizes, int n_in,
                              void* d_out, int out_size, void* d_ws, size_t ws_size,
                              hipStream_t stream) {
    const float* x    = (const float*)d_in[0];
    const int*   ei   = (const int*)d_in[1];
    const int*   batch= (const int*)d_in[2];
    const float* w0 = (const float*)d_in[3];  const float* b0 = (const float*)d_in[4];
    const float* w1 = (const float*)d_in[5];  const float* b1 = (const float*)d_in[6];
    const float* w2 = (const float*)d_in[7];  const float* b2 = (const float*)d_in[8];
    const float* g0 = (const float*)d_in[9];  const float* be0 = (const float*)d_in[10];
    const float* m0 = (const float*)d_in[11]; const float* v0  = (const float*)d_in[12];
    const float* g1 = (const float*)d_in[13]; const float* be1 = (const float*)d_in[14];
    const float* m1 = (const float*)d_in[15]; const float* v1  = (const float*)d_in[16];
    const float* g2 = (const float*)d_in[17]; const float* be2 = (const float*)d_in[18];
    const float* m2 = (const float*)d_in[19]; const float* v2  = (const float*)d_in[20];
    const float* l1w = (const float*)d_in[21]; const float* l1b = (const float*)d_in[22];
    const float* l2w = (const float*)d_in[23]; const float* l2b = (const float*)d_in[24];

    const int n   = in_sizes[2];           // 100000 nodes
    const int e   = in_sizes[1] / 2;       // 1.6M edges
    const int fin = in_sizes[0] / n;       // 128
    const int G   = out_size / 2;          // 256 graphs

    const int* srcI = ei;
    const int* dstI = ei + e;

    // Workspace carve-up
    char* ws = (char*)d_ws;
    size_t off = 0;
    auto carve = [&](size_t bytes) -> void* {
        void* p = ws + off;
        off = (off + bytes + 255) & ~(size_t)255;
        return p;
    };
    float* dinv   = (float*)carve((size_t)n * 4);
    float* norm   = (float*)carve((size_t)e * 4);
    float* t      = (float*)carve((size_t)n * H_DIM * 4);
    float* agg    = (float*)carve((size_t)n * H_DIM * 4);
    float* h      = (float*)carve((size_t)n * H_DIM * 4);
    float* pooled = (float*)carve((size_t)G * H_DIM * 4);
    float* mid    = (float*)carve((size_t)G * 32 * 4);
    (void)ws_size; (void)n_in;

    const int TB = 256;

    // Symmetric normalization coefficients (computed once, reused 3 layers)
    gcn_deg_init  <<<cdiv_i(n, TB), TB, 0, stream>>>(dinv, n);
    gcn_deg_count <<<cdiv_i(e, TB), TB, 0, stream>>>(dstI, dinv, e);
    gcn_deg_rsqrt <<<cdiv_i(n, TB), TB, 0, stream>>>(dinv, n);
    gcn_edge_norm <<<cdiv_i(e, TB), TB, 0, stream>>>(srcI, dstI, dinv, norm, e);

    const int gemm_blocks = cdiv_i(n, ROWS_PER_BLOCK);
    const size_t smem128 = (size_t)ROWS_PER_BLOCK * (128 + 4) * 4;  // 132 KB
    const size_t smem64  = (size_t)ROWS_PER_BLOCK * (64 + 4) * 4;   // 68 KB
    (void)hipFuncSetAttribute(reinterpret_cast<const void*>(&gcn_gemm_wmma_k<128>),
                              hipFuncAttributeMaxDynamicSharedMemorySize, (int)smem128);
    (void)hipFuncSetAttribute(reinterpret_cast<const void*>(&gcn_gemm_wmma_k<64>),
                              hipFuncAttributeMaxDynamicSharedMemorySize, (int)smem64);

    const int nEl = n * H_DIM;
    const int n16 = n * 16;
    const int e16 = e * 16;
    (void)fin;  // reference fin == 128

    // --- Layer 0: t = x @ w0 ; agg = norm-scatter(t) ; h = relu(bn(agg + b0)) ---
    gcn_gemm_wmma_k<128><<<gemm_blocks, 128, smem128, stream>>>(x, w0, t, n);
    gcn_selfloop_init <<<cdiv_i(n16, TB), TB, 0, stream>>>(t, dinv, agg, n);
    gcn_edge_scatter  <<<cdiv_i(e16, TB), TB, 0, stream>>>(srcI, dstI, norm, t, agg, e);
    gcn_post_bn_relu  <<<cdiv_i(nEl, TB), TB, 0, stream>>>(agg, b0, g0, be0, m0, v0, h, n);

    // --- Layer 1 ---
    gcn_gemm_wmma_k<64><<<gemm_blocks, 128, smem64, stream>>>(h, w1, t, n);
    gcn_selfloop_init <<<cdiv_i(n16, TB), TB, 0, stream>>>(t, dinv, agg, n);
    gcn_edge_scatter  <<<cdiv_i(e16, TB), TB, 0, stream>>>(srcI, dstI, norm, t, agg, e);
    gcn_post_bn_relu  <<<cdiv_i(nEl, TB), TB, 0, stream>>>(agg, b1, g1, be1, m1, v1, h, n);

    // --- Layer 2 ---
    gcn_gemm_wmma_k<64><<<gemm_blocks, 128, smem64, stream>>>(h, w2, t, n);
    gcn_selfloop_init <<<cdiv_i(n16, TB), TB, 0, stream>>>(t, dinv, agg, n);
    gcn_edge_scatter  <<<cdiv_i(e16, TB), TB, 0, stream>>>(srcI, dstI, norm, t, agg, e);
    gcn_post_bn_relu  <<<cdiv_i(nEl, TB), TB, 0, stream>>>(agg, b2, g2, be2, m2, v2, h, n);

    // --- Pool + MLP head ---
    gcn_zero <<<cdiv_i(G * H_DIM, TB), TB, 0, stream>>>(pooled, G * H_DIM);
    gcn_pool <<<cdiv_i(n16, TB), TB, 0, stream>>>(h, batch, pooled, n);
    gcn_mlp1 <<<cdiv_i(G * 32, TB), TB, 0, stream>>>(pooled, l1w, l1b, mid, G);
    gcn_mlp2 <<<cdiv_i(G * 2, TB), TB, 0, stream>>>(mid, l2w, l2b, (float*)d_out, G);
}